// modDeepGCN_62998580298151
// MI455X (gfx1250) — compile-verified
//
#include <hip/hip_runtime.h>
#include <hip/hip_bf16.h>
#include <math.h>

// ---------------------------------------------------------------------------
// modDeepGCN for MI455X (gfx1250, wave32).
// GEMMs: V_WMMA_F32_16X16X4_F32, B staged transposed+zero-padded in LDS so the
// inner loop is branch-free (global_load_b64 for A, ds_load_b64 for B).
// Edge scatter / aggregation: f32 global atomics; feature buffers (102MB)
// fit in the 192MB L2 so the irregular traffic stays on-die.
// ---------------------------------------------------------------------------

typedef float v2f __attribute__((ext_vector_type(2)));
typedef float v8f __attribute__((ext_vector_type(8)));

#define LDC 64   // common row stride for all feature buffers

// ---------------------------- WMMA GEMM ------------------------------------
// C[M, NT*16] = A[M,K] @ B[K, fo] (+bias), fo <= NT*16, M % 16 == 0.
// One wave computes a 16-row tile across NT column tiles of 16.
// W is staged in LDS transposed: WT[col][k], col padded to NT*16 with zeros,
// row stride KP = K+4 dwords (bank-conflict-free for wave32 x b64 reads).
template <int NT, int K>
__global__ void gemm_wmma_f32(const float* __restrict__ A, int lda,
                              const float* __restrict__ Bw,   // [K, fo]
                              const float* __restrict__ bias, // may be null
                              float* __restrict__ C, int ldc,
                              int M, int fo)
{
    constexpr int NC = NT * 16;
    constexpr int KP = K + 4;
    extern __shared__ float WT[];            // [NC][KP]

    // cooperative transpose + zero-pad of W into LDS
    for (int idx = threadIdx.x; idx < NC * KP; idx += blockDim.x) {
        const int col = idx / KP;
        const int k   = idx - col * KP;
        float v = 0.0f;
        if (k < K && col < fo) v = Bw[k * fo + col];
        WT[idx] = v;
    }
    __syncthreads();

    const int wave = threadIdx.x >> 5;       // 8 waves / block
    const int lane = threadIdx.x & 31;
    const int row0 = (blockIdx.x * 8 + wave) * 16;
    if (row0 >= M) return;                   // wave-uniform (tail block only)

    const int m  = lane & 15;                // A row / B col within tile
    const int kh = lane >> 4;                // 0 -> K{0,1}, 1 -> K{2,3}
    const float* ar = A + (long)(row0 + m) * lda + 2 * kh;
    const float* bt = WT + m * KP + 2 * kh;

    v8f zero = {};
    v8f acc[NT];
#pragma unroll
    for (int t = 0; t < NT; ++t) acc[t] = zero;

#pragma unroll 4
    for (int k = 0; k < K; k += 4) {
        const v2f a = *(const v2f*)(ar + k);
#pragma unroll
        for (int t = 0; t < NT; ++t) {
            const v2f b = *(const v2f*)(bt + t * 16 * KP + k);
            acc[t] = __builtin_amdgcn_wmma_f32_16x16x4_f32(
                false, a, false, b, (short)0, acc[t], false, false);
        }
    }

    // D layout: VGPR j -> M = j (+8 for lanes 16..31), N = lane&15
    const int cm = kh * 8;
    float bv[NT];
#pragma unroll
    for (int t = 0; t < NT; ++t) bv[t] = bias ? bias[t * 16 + m] : 0.0f;

#pragma unroll
    for (int j = 0; j < 8; ++j) {
        float* crow = C + (long)(row0 + cm + j) * ldc + m;
#pragma unroll
        for (int t = 0; t < NT; ++t)
            crow[t * 16] = acc[t][j] + bv[t];
    }
}

// ---------------------------- degree / norm --------------------------------
__global__ void k_deg_init(float* dis, int N)
{
    int i = blockIdx.x * blockDim.x + threadIdx.x;
    if (i < N) dis[i] = 1.0f;          // self loop
}
__global__ void k_deg_accum(const int* __restrict__ dst, float* dis, long E)
{
    long e = (long)blockIdx.x * blockDim.x + threadIdx.x;
    if (e < E) atomicAdd(&dis[dst[e]], 1.0f);
}
__global__ void k_rsqrt(float* dis, int N)
{
    int i = blockIdx.x * blockDim.x + threadIdx.x;
    if (i < N) dis[i] = rsqrtf(dis[i]);
}

// ---------------------------- scatter (A_hat @ H) --------------------------
__global__ void k_bias_init(float* out, const float* __restrict__ bias, long N, int fo)
{
    long t = (long)blockIdx.x * blockDim.x + threadIdx.x;
    if (t >= N * LDC) return;
    int c = (int)(t & (LDC - 1));
    out[t] = (c < fo) ? bias[c] : 0.0f;
}

__global__ void k_scatter(const float* __restrict__ tmp,
                          const int* __restrict__ srcA,
                          const int* __restrict__ dstA,
                          const float* __restrict__ dis,
                          float* __restrict__ out,
                          long E, long N, int chunks)
{
    long tid = (long)blockIdx.x * blockDim.x + threadIdx.x;
    long total = (E + N) * chunks;
    if (tid >= total) return;
    long e  = tid / chunks;
    int  c4 = (int)(tid - e * chunks) * 4;
    int s, d; float nrm;
    if (e < E) {
        // keep the index stream ahead of the atomic pipeline
        if (e + 16384 < E) {
            __builtin_prefetch(&srcA[e + 16384], 0, 1);
            __builtin_prefetch(&dstA[e + 16384], 0, 1);
        }
        s = srcA[e]; d = dstA[e]; nrm = dis[s] * dis[d];
    } else {
        s = d = (int)(e - E);
        float w = dis[s]; nrm = w * w;
    }
    const float4 v = *(const float4*)(tmp + (long)s * LDC + c4);
    float* o = out + (long)d * LDC + c4;
    atomicAdd(o + 0, nrm * v.x);
    atomicAdd(o + 1, nrm * v.y);
    atomicAdd(o + 2, nrm * v.z);
    atomicAdd(o + 3, nrm * v.w);
}

// ---------------------------- BatchNorm ------------------------------------
__global__ void k_zero(float* p, int n)
{
    int i = blockIdx.x * blockDim.x + threadIdx.x;
    if (i < n) p[i] = 0.0f;
}

__global__ void k_bn_stats(const float* __restrict__ h, long N, int fo,
                           float* __restrict__ sum, float* __restrict__ sumsq)
{
    __shared__ float s1[256], s2[256];
    const int c  = threadIdx.x & 63;
    const int rs = threadIdx.x >> 6;          // 0..3
    const long start = (long)blockIdx.x * 1024;
    float a = 0.0f, b = 0.0f;
    if (c < fo) {
        long end = start + 1024; if (end > N) end = N;
        for (long r = start + rs; r < end; r += 4) {
            float v = h[r * LDC + c];
            a += v; b += v * v;
        }
    }
    s1[threadIdx.x] = a; s2[threadIdx.x] = b;
    __syncthreads();
    if (rs == 0 && c < fo) {
        a = s1[c] + s1[c + 64] + s1[c + 128] + s1[c + 192];
        b = s2[c] + s2[c + 64] + s2[c + 128] + s2[c + 192];
        atomicAdd(&sum[c], a);
        atomicAdd(&sumsq[c], b);
    }
}

__global__ void k_bn_final(const float* __restrict__ sum, const float* __restrict__ sumsq,
                           const float* __restrict__ g, const float* __restrict__ be,
                           float* __restrict__ scale, float* __restrict__ shift,
                           long N, int fo)
{
    int c = threadIdx.x;
    if (c >= fo) return;
    float invn = 1.0f / (float)N;
    float mu  = sum[c] * invn;
    float var = sumsq[c] * invn - mu * mu;
    float sc  = g[c] * rsqrtf(var + 1e-5f);
    scale[c] = sc;
    shift[c] = be[c] - mu * sc;
}

__global__ void k_bn_apply(float* __restrict__ h, const float* __restrict__ scale,
                           const float* __restrict__ shift, long N, int fo, int relu)
{
    long t = (long)blockIdx.x * blockDim.x + threadIdx.x;
    if (t >= N * LDC) return;
    int c = (int)(t & (LDC - 1));
    if (c >= fo) return;
    float v = h[t] * scale[c] + shift[c];
    if (relu) v = fmaxf(v, 0.0f);
    h[t] = v;
}

// ---------------------------- aggregation ----------------------------------
__device__ __forceinline__ void atomicMaxF(float* addr, float val)
{
    int* ai = (int*)addr;
    int old = __float_as_int(*addr);
    while (__int_as_float(old) < val) {
        int assumed = old;
        old = atomicCAS(ai, assumed, __float_as_int(val));
        if (old == assumed) break;
    }
}
__device__ __forceinline__ void atomicMinF(float* addr, float val)
{
    int* ai = (int*)addr;
    int old = __float_as_int(*addr);
    while (__int_as_float(old) > val) {
        int assumed = old;
        old = atomicCAS(ai, assumed, __float_as_int(val));
        if (old == assumed) break;
    }
}

__global__ void k_agg_init(float* cnt, float* sm, float* sm2, float* mn, float* mx, int S)
{
    int t = blockIdx.x * blockDim.x + threadIdx.x;
    if (t >= S * 64) return;
    sm[t]  = 0.0f;
    sm2[t] = 0.0f;
    mn[t]  =  INFINITY;
    mx[t]  = -INFINITY;
    if ((t & 63) == 0) cnt[t >> 6] = 0.0f;
}

__global__ void k_agg_node(const float* __restrict__ h, const int* __restrict__ seg,
                           float* cnt, float* sm, float* sm2, float* mn, float* mx,
                           long N)
{
    long t = (long)blockIdx.x * blockDim.x + threadIdx.x;
    if (t >= N * 64) return;
    long n = t >> 6;
    int  c = (int)(t & 63);
    int  s = seg[n];
    float v = h[n * LDC + c];
    long  o = (long)s * 64 + c;
    atomicAdd(&sm[o],  v);
    atomicAdd(&sm2[o], v * v);
    atomicMinF(&mn[o], v);
    atomicMaxF(&mx[o], v);
    if (c == 0) atomicAdd(&cnt[s], 1.0f);
}

__global__ void k_agg_build(const float* cnt, const float* sm, const float* sm2,
                            const float* mn, const float* mx,
                            float* __restrict__ agg, int S)
{
    int t = blockIdx.x * blockDim.x + threadIdx.x;
    if (t >= S * 64) return;
    int s = t >> 6, c = t & 63;
    float cn   = fmaxf(cnt[s], 1.0f);
    float mean = sm[t] / cn;
    float var  = sm2[t] / cn - mean * mean;
    float sd   = sqrtf(fmaxf(var, 0.0f) + 1e-5f);
    float* row = agg + (long)s * 256;
    row[c]       = mean;
    row[64 + c]  = mn[t];
    row[128 + c] = mx[t];
    row[192 + c] = sd;
}

// ---------------------------- final pack -----------------------------------
// y[b,c,j,i] = proj[clip(offs[b] + (i*14+j))] [c]  masked by pos < num_sp[b]
__global__ void k_pack(const float* __restrict__ proj, const int* __restrict__ num_sp,
                       float* __restrict__ out, int Bimg, int S)
{
    int t = blockIdx.x * blockDim.x + threadIdx.x;
    int total = Bimg * 64 * 196;
    if (t >= total) return;
    int i = t % 14;
    int j = (t / 14) % 14;
    int c = (t / 196) % 64;
    int b = t / (196 * 64);
    int pos = i * 14 + j;
    int offs = 0;
    for (int k = 0; k < b; ++k) offs += num_sp[k];
    float v = 0.0f;
    if (pos < num_sp[b]) {
        int idx = offs + pos;
        idx = idx < 0 ? 0 : (idx > S - 1 ? S - 1 : idx);
        v = proj[(long)idx * 64 + c];
    }
    out[t] = v;
}

// ---------------------------- orchestration --------------------------------
static inline unsigned blocks_for(long n, int bs) { return (unsigned)((n + bs - 1) / bs); }

#define LAUNCH_GEMM(NT, K, A, lda, Bw, bias, Cp, M, fo)                           \
    gemm_wmma_f32<NT, K><<<blocks_for(((M) + 15) / 16, 8), 256,                   \
                           (NT * 16) * ((K) + 4) * sizeof(float), stream>>>(      \
        A, lda, Bw, bias, Cp, LDC, M, fo)

extern "C" void kernel_launch(void* const* d_in, const int* in_sizes, int n_in,
                              void* d_out, int out_size, void* d_ws, size_t ws_size,
                              hipStream_t stream)
{
    const float* x          = (const float*)d_in[0];   // [N,24]
    const int*   edge_index = (const int*)  d_in[1];   // [2,E]
    const int*   batch_idx  = (const int*)  d_in[2];   // [N]
    const int*   num_sp     = (const int*)  d_in[3];   // [B]
    const float* W[5], *bb[5], *gg[5], *be[5];
    for (int i = 0; i < 5; ++i) {
        W[i]  = (const float*)d_in[4 + 4 * i + 0];
        bb[i] = (const float*)d_in[4 + 4 * i + 1];
        gg[i] = (const float*)d_in[4 + 4 * i + 2];
        be[i] = (const float*)d_in[4 + 4 * i + 3];
    }
    const float* Wp = (const float*)d_in[24];          // [256,64]
    const float* bp = (const float*)d_in[25];          // [64]
    float* out = (float*)d_out;

    const long N    = in_sizes[2];
    const long E    = in_sizes[1] / 2;
    const int  Bimg = in_sizes[3];
    const int  S    = Bimg * 150;                      // NUM_SP = 150 (reference)
    const int  DIMS[6] = {24, 8, 16, 32, 64, 64};

    const int* srcA = edge_index;
    const int* dstA = edge_index + E;

    // ---- workspace carve-up (floats) ----
    float* ws  = (float*)d_ws;
    size_t off = 0;
    float* dis   = ws + off; off += (size_t)N;
    float* buf0  = ws + off; off += (size_t)N * LDC;
    float* buf1  = ws + off; off += (size_t)N * LDC;
    float* buf2  = ws + off; off += (size_t)N * LDC;
    float* bnsum = ws + off; off += 64;
    float* bnsq  = ws + off; off += 64;
    float* bnsc  = ws + off; off += 64;
    float* bnsh  = ws + off; off += 64;
    float* cnt   = ws + off; off += (size_t)S;
    float* sm    = ws + off; off += (size_t)S * 64;
    float* sm2   = ws + off; off += (size_t)S * 64;
    float* mn    = ws + off; off += (size_t)S * 64;
    float* mx    = ws + off; off += (size_t)S * 64;
    float* agg   = ws + off; off += (size_t)S * 256;
    float* proj  = ws + off; off += (size_t)S * 64;
    (void)ws_size;

    float* bufs[3] = {buf0, buf1, buf2};

    // ---- degree / symmetric normalization ----
    k_deg_init <<<blocks_for(N, 256), 256, 0, stream>>>(dis, (int)N);
    k_deg_accum<<<blocks_for(E, 256), 256, 0, stream>>>(dstA, dis, E);
    k_rsqrt    <<<blocks_for(N, 256), 256, 0, stream>>>(dis, (int)N);

    // ---- 5 GCN layers (buffer rotation: in / gemm-tmp / scatter-out) ----
    const int inI[5]  = {-1, 1, 0, 2, 1};
    const int tmpI[5] = { 0, 2, 1, 0, 2};
    const int outI[5] = { 1, 0, 2, 1, 0};

    for (int L = 0; L < 5; ++L) {
        const int fi = DIMS[L], fo = DIMS[L + 1];
        const float* hin = (inI[L] < 0) ? x : bufs[inI[L]];
        const int lda    = (inI[L] < 0) ? fi : LDC;
        float* tmp = bufs[tmpI[L]];
        float* o   = bufs[outI[L]];

        switch (L) {  // K (=fi) and NT are compile-time
        case 0: LAUNCH_GEMM(1, 24, hin, lda, W[L], nullptr, tmp, (int)N, fo); break;
        case 1: LAUNCH_GEMM(1,  8, hin, lda, W[L], nullptr, tmp, (int)N, fo); break;
        case 2: LAUNCH_GEMM(2, 16, hin, lda, W[L], nullptr, tmp, (int)N, fo); break;
        case 3: LAUNCH_GEMM(4, 32, hin, lda, W[L], nullptr, tmp, (int)N, fo); break;
        case 4: LAUNCH_GEMM(4, 64, hin, lda, W[L], nullptr, tmp, (int)N, fo); break;
        }

        k_bias_init<<<blocks_for(N * LDC, 256), 256, 0, stream>>>(o, bb[L], N, fo);

        const int chunks = fo / 4;
        k_scatter<<<blocks_for((E + N) * chunks, 256), 256, 0, stream>>>(
            tmp, srcA, dstA, dis, o, E, N, chunks);

        k_zero    <<<1, 128, 0, stream>>>(bnsum, 128);   // bnsum + bnsq (contiguous)
        k_bn_stats<<<blocks_for(N, 1024), 256, 0, stream>>>(o, N, fo, bnsum, bnsq);
        k_bn_final<<<1, 64, 0, stream>>>(bnsum, bnsq, gg[L], be[L], bnsc, bnsh, N, fo);
        k_bn_apply<<<blocks_for(N * LDC, 256), 256, 0, stream>>>(o, bnsc, bnsh, N, fo, (L < 4) ? 1 : 0);
    }
    float* hfin = bufs[outI[4]];  // buf0

    // ---- segment multi-aggregation (mean / min / max / std) ----
    k_agg_init <<<blocks_for((long)S * 64, 256), 256, 0, stream>>>(cnt, sm, sm2, mn, mx, S);
    k_agg_node <<<blocks_for(N * 64, 256), 256, 0, stream>>>(hfin, batch_idx, cnt, sm, sm2, mn, mx, N);
    k_agg_build<<<blocks_for((long)S * 64, 256), 256, 0, stream>>>(cnt, sm, sm2, mn, mx, agg, S);

    // ---- projection: [S,256] @ [256,64] + bp (WMMA) ----
    LAUNCH_GEMM(4, 256, agg, 256, Wp, bp, proj, S, 64);

    // ---- pack into [B, 64, 14, 14] ----
    k_pack<<<blocks_for((long)Bimg * 64 * 196, 256), 256, 0, stream>>>(proj, num_sp, out, Bimg, S);
}